// GraphRespiratory_75788992905528
// MI455X (gfx1250) — compile-verified
//
#include <hip/hip_runtime.h>
#include <hip/hip_bf16.h>
#include <math.h>

// ---------------------------------------------------------------------------
// Types for CDNA5 WMMA fp32 16x16x4
// ---------------------------------------------------------------------------
typedef __attribute__((ext_vector_type(2))) float v2f;
typedef __attribute__((ext_vector_type(8))) float v8f;

__device__ __forceinline__ v8f wmma4(v2f a, v2f b, v8f c) {
  // D = A(16x4) x B(4x16) + C(16x16), fp32 in / fp32 out
  return __builtin_amdgcn_wmma_f32_16x16x4_f32(false, a, false, b, (short)0, c,
                                               false, false);
}

__device__ __forceinline__ v2f ld2(const float* p) {
  return *reinterpret_cast<const v2f*>(p);  // 8B aligned at call sites
}

__device__ __forceinline__ float sigm(float x) { return 1.0f / (1.0f + expf(-x)); }

// ---------------------------------------------------------------------------
// Kernel 1: gx[sd] = X @ Wih[sd]^T + bih[sd]   (M=512, N=768, K=256) x 6
// One wave per 16x16 tile.
// ---------------------------------------------------------------------------
struct GxArgs {
  const float* X;        // (512,256)
  const float* wih[6];   // (768,256) each
  const float* bih[6];   // (768,)
  float* gx;             // (6,512,768)
};

__global__ void gx_kernel(GxArgs a) {
  const int wave = (blockIdx.x * blockDim.x + threadIdx.x) >> 5;
  const int lane = threadIdx.x & 31;
  const int half = lane >> 4;      // 0: K+0/1, 1: K+2/3
  const int l    = lane & 15;
  const int tilesPerSd = 32 * 48;  // (512/16)*(768/16)
  const int sd = wave / tilesPerSd;
  if (sd >= 6) return;
  const int t  = wave - sd * tilesPerSd;
  const int mt = t / 48, nt = t - mt * 48;
  const int m0 = mt * 16, n0 = nt * 16;

  const float* X = a.X;
  const float* W = a.wih[sd];
  float* gx = a.gx + (size_t)sd * 512 * 768;

  v8f acc;
  const float bv = a.bih[sd][n0 + l];  // C layout: col = lane%16 for both halves
  #pragma unroll
  for (int v = 0; v < 8; ++v) acc[v] = bv;

  #pragma unroll 4
  for (int k = 0; k < 256; k += 4) {
    const int ka = k + half * 2;
    v2f af = ld2(X + (size_t)(m0 + l) * 256 + ka);  // A: row=l, K=ka..ka+1
    v2f bf = ld2(W + (size_t)(n0 + l) * 256 + ka);  // B[k][n] = Wih[n][k]
    acc = wmma4(af, bf, acc);
  }
  #pragma unroll
  for (int v = 0; v < 8; ++v)
    gx[(size_t)(m0 + v + 8 * half) * 768 + n0 + l] = acc[v];
}

// ---------------------------------------------------------------------------
// Kernel 2: max|w| for fq() scales (9 tensors), one block per tensor
// ---------------------------------------------------------------------------
struct MaxArgs {
  const float* w[9];
  int len[9];
  unsigned int* out;  // 9 slots, float bits (non-negative -> uint order ok)
};

__global__ void maxabs_kernel(MaxArgs a) {
  __shared__ float red[256];
  const int b = blockIdx.x;
  const float* w = a.w[b];
  const int n = a.len[b];
  float m = 0.0f;
  for (int i = threadIdx.x; i < n; i += 256) m = fmaxf(m, fabsf(w[i]));
  red[threadIdx.x] = m;
  __syncthreads();
  for (int s = 128; s > 0; s >>= 1) {
    if (threadIdx.x < s) red[threadIdx.x] = fmaxf(red[threadIdx.x], red[threadIdx.x + s]);
    __syncthreads();
  }
  if (threadIdx.x == 0) a.out[b] = __float_as_uint(red[0]);
}

// ---------------------------------------------------------------------------
// Kernel 3: sequential GRU scan.
// Anchors (batch rows) are fully independent in the recurrence, so we split
// rows across workgroups: one block per (scale,dir,16-row tile) = 10 blocks.
// 512 threads = 16 waves; wave w owns hidden-column group w (16 cols) and
// computes its r/z/n WMMA tiles (cols j, 256+j, 512+j) so the gate update
// needs no gh staging. h (16x256) lives in LDS. fp32 WMMA for reference-grade
// numerics over the 512-step chain.
// ---------------------------------------------------------------------------
struct ScanArgs {
  const float* whh[6];   // (768,256)
  const float* bhh[6];   // (768,)
  const float* gx;       // (6,512,768)
  const float* tp;       // (512,)
  const float* audio_len;// (1,)
  const float* anch;     // (70,2)
  float* hfin;           // (6,48,256)
};

__global__ __launch_bounds__(512) void gru_scan_kernel(ScanArgs a) {
  __shared__ float hsm[16 * 256];  // 16KB: this block's 16-row h slab

  // block -> (sd, mtile): s0f, s0b, s1f x3, s1b x3, s2f, s2b
  const int b = blockIdx.x;
  int sd, mtile;
  if (b < 2)      { sd = b;              mtile = 0; }
  else if (b < 8) { sd = 2 + (b - 2) / 3; mtile = (b - 2) % 3; }
  else            { sd = 4 + (b - 8);    mtile = 0; }

  const int s    = sd >> 1;
  const bool rev = (sd & 1) != 0;
  const int I    = (s == 1) ? 40 : 15;
  const int rowoff = (s == 0) ? 0 : ((s == 1) ? 15 : 55);
  const int rbase  = mtile * 16;  // first batch row handled by this block

  const int wave = threadIdx.x >> 5;  // 0..15 = hidden column group
  const int lane = threadIdx.x & 31;
  const int half = lane >> 4;
  const int l    = lane & 15;

  const float* whh = a.whh[sd];
  const float* bhh = a.bhh[sd];
  const float* gx  = a.gx + (size_t)sd * 512 * 768;
  const float  Lv  = a.audio_len[0];

  const int jr = wave * 16 + l;  // r-gate column (also hidden index d)
  const int jz = 256 + jr;
  const int jn = 512 + jr;
  const float b_r = bhh[jr], b_z = bhh[jz], b_n = bhh[jn];

  // per-lane anchor bounds for the eight rows this lane updates (v + 8*half pattern
  // handled below; bounds fetched per element but L2/L0-hot)

  // h0 = 0
  for (int idx = threadIdx.x; idx < 16 * 256; idx += 512) hsm[idx] = 0.0f;
  __syncthreads();

  for (int step = 0; step < 512; ++step) {
    const int t = rev ? (511 - step) : step;
    const float tpos = a.tp[t] * Lv;

    v8f accR, accZ, accN;
    #pragma unroll
    for (int v = 0; v < 8; ++v) { accR[v] = b_r; accZ[v] = b_z; accN[v] = b_n; }

    // gh tiles: gh[i][j] = sum_k h[i][k] * whh[j][k]
    #pragma unroll 4
    for (int k = 0; k < 256; k += 4) {
      const int ka = k + half * 2;
      v2f br = ld2(whh + (size_t)jr * 256 + ka);
      v2f bz = ld2(whh + (size_t)jz * 256 + ka);
      v2f bn = ld2(whh + (size_t)jn * 256 + ka);
      v2f af = ld2(&hsm[(size_t)l * 256 + ka]);  // A: row=l, K=ka..ka+1
      accR = wmma4(af, br, accR);
      accZ = wmma4(af, bz, accZ);
      accN = wmma4(af, bn, accN);
    }

    // prefetch next step's gx row while gates compute
    const int tn = rev ? (t - 1) : (t + 1);
    if (tn >= 0 && tn < 512) __builtin_prefetch(gx + (size_t)tn * 768 + jr, 0, 0);

    __syncthreads();  // everyone finished reading h via WMMA A-frags

    const float gxr = gx[(size_t)t * 768 + jr];
    const float gxz = gx[(size_t)t * 768 + jz];
    const float gxn = gx[(size_t)t * 768 + jn];

    #pragma unroll
    for (int v = 0; v < 8; ++v) {
      const int lrow = v + 8 * half;       // local row in this 16-row slab
      const int irow = rbase + lrow;       // batch row within scale
      const float r = sigm(gxr + accR[v]);
      const float z = sigm(gxz + accZ[v]);
      const float n = tanhf(gxn + r * accN[v]);
      const float hold = hsm[(size_t)lrow * 256 + jr];
      const float hnew = (1.0f - z) * n + z * hold;
      bool m = false;
      if (irow < I) {
        const float st = a.anch[(rowoff + irow) * 2];
        const float en = a.anch[(rowoff + irow) * 2 + 1];
        m = (tpos >= st) && (tpos <= en);
      }
      hsm[(size_t)lrow * 256 + jr] = m ? hnew : hold;
    }
    __syncthreads();
  }

  // write this block's 16-row slab of the final h
  float* hf = a.hfin + (size_t)sd * 48 * 256 + (size_t)rbase * 256;
  for (int idx = threadIdx.x; idx < 16 * 256; idx += 512) hf[idx] = hsm[idx];
}

// ---------------------------------------------------------------------------
// Kernel 4: MLP head, one block per anchor (70 blocks, 256 threads)
// ---------------------------------------------------------------------------
struct HeadArgs {
  const float* hfin;
  const float* w1[3]; const float* w2[3]; const float* w3[3];
  const float* sw[3]; const float* ew[3];
  const float* bih_f[3]; const float* bhh_f[3];
  const float* bih_b[3]; const float* bhh_b[3];
  const unsigned int* scales;  // 9
  const float* node_pred;      // (512,5)
  const float* tp;             // (512,)
  const float* audio_len;      // (1,)
  const float* anch;           // (70,2)
  float* out;                  // 490 floats
};

__device__ __forceinline__ float fq_w(float w, float q) {
  return fminf(fmaxf(rintf(w / q), -128.0f), 127.0f) * q;
}

__global__ void head_kernel(HeadArgs a) {
  __shared__ float feat[515];
  __shared__ float h1s[256];
  __shared__ float h2s[256];
  __shared__ float outv[168];
  __shared__ int flag;

  const int bid = blockIdx.x;
  int s, i;
  if (bid < 15)      { s = 0; i = bid; }
  else if (bid < 55) { s = 1; i = bid - 15; }
  else               { s = 2; i = bid - 55; }
  const int rowoff = (s == 0) ? 0 : ((s == 1) ? 15 : 55);
  const int grow = rowoff + i;
  const int tid = threadIdx.x;
  const float Lv = a.audio_len[0];
  const float st = a.anch[grow * 2], en = a.anch[grow * 2 + 1];

  if (tid == 0) flag = 0;
  __syncthreads();
  for (int t = tid; t < 512; t += 256) {
    const float tpos = a.tp[t] * Lv;
    if (tpos >= st && tpos <= en) atomicOr(&flag, 1);
  }
  __syncthreads();
  const bool empty = (flag == 0);

  const float* hff = a.hfin + (size_t)(s * 2) * 48 * 256 + (size_t)i * 256;
  const float* hfb = a.hfin + (size_t)(s * 2 + 1) * 48 * 256 + (size_t)i * 256;
  {
    const int d = tid;  // 256 threads, one hidden dim each
    float vf, vb;
    if (empty) {
      float r = sigm(a.bih_f[s][d] + a.bhh_f[s][d]);
      float z = sigm(a.bih_f[s][256 + d] + a.bhh_f[s][256 + d]);
      float n = tanhf(a.bih_f[s][512 + d] + r * a.bhh_f[s][512 + d]);
      vf = (1.0f - z) * n;
      float rb = sigm(a.bih_b[s][d] + a.bhh_b[s][d]);
      float zb = sigm(a.bih_b[s][256 + d] + a.bhh_b[s][256 + d]);
      float nb = tanhf(a.bih_b[s][512 + d] + rb * a.bhh_b[s][512 + d]);
      vb = (1.0f - zb) * nb;
    } else { vf = hff[d]; vb = hfb[d]; }
    feat[d] = vf;
    feat[256 + d] = vb;
  }
  if (tid == 0) {
    const float* npd = a.node_pred + (size_t)i * 5;
    float mx = npd[0];
    for (int c = 1; c < 5; ++c) mx = fmaxf(mx, npd[c]);
    float se = 0.0f;
    for (int c = 0; c < 5; ++c) se += expf(npd[c] - mx);
    feat[512] = expf(npd[0] - mx) / se;   // abnormal prob, class 0
    feat[513] = (st + en) / (2.0f * Lv);
    feat[514] = (en - st) / Lv;
  }
  __syncthreads();

  const float q1 = __uint_as_float(a.scales[s * 3 + 0]) / 127.0f + 1e-12f;
  const float q2 = __uint_as_float(a.scales[s * 3 + 1]) / 127.0f + 1e-12f;
  const float q3 = __uint_as_float(a.scales[s * 3 + 2]) / 127.0f + 1e-12f;

  {
    const float* w1 = a.w1[s] + (size_t)tid * 515;
    float acc = 0.0f;
    for (int k = 0; k < 515; ++k) acc += feat[k] * fq_w(w1[k], q1);
    h1s[tid] = fmaxf(acc, 0.0f);
  }
  __syncthreads();
  {
    const float* w2 = a.w2[s] + (size_t)tid * 256;
    float acc = 0.0f;
    for (int k = 0; k < 256; ++k) acc += h1s[k] * fq_w(w2[k], q2);
    h2s[tid] = fmaxf(acc, 0.0f);
  }
  __syncthreads();
  const int bins = (s == 1) ? 60 : 80;
  const int nout = 2 * bins + 5;
  if (tid < nout) {
    const float* w3 = a.w3[s] + (size_t)tid * 256;
    float acc = 0.0f;
    for (int k = 0; k < 256; ++k) acc += h2s[k] * fq_w(w3[k], q3);
    outv[tid] = acc;
  }
  __syncthreads();
  if (tid == 0) {
    float mx = -1e30f;
    for (int k = 0; k < bins; ++k) mx = fmaxf(mx, outv[k]);
    float se = 0.0f, wsum = 0.0f;
    for (int k = 0; k < bins; ++k) { float e = expf(outv[k] - mx); se += e; wsum += e * a.sw[s][k]; }
    const float so = wsum / se;
    mx = -1e30f;
    for (int k = 0; k < bins; ++k) mx = fmaxf(mx, outv[bins + k]);
    se = 0.0f; wsum = 0.0f;
    for (int k = 0; k < bins; ++k) { float e = expf(outv[bins + k] - mx); se += e; wsum += e * a.ew[s][k]; }
    const float eo = wsum / se;
    a.out[grow * 2 + 0] = fminf(fmaxf(st + so, 0.0f), Lv);  // bounds
    a.out[grow * 2 + 1] = fminf(fmaxf(en + eo, 0.0f), Lv);
    a.out[140 + grow]   = outv[2 * bins];                   // conf
  }
  if (tid < 4) a.out[210 + grow * 4 + tid] = outv[2 * bins + 1 + tid];  // class logits
}

// ---------------------------------------------------------------------------
// Host launcher
// ---------------------------------------------------------------------------
extern "C" void kernel_launch(void* const* d_in, const int* in_sizes, int n_in,
                              void* d_out, int out_size, void* d_ws, size_t ws_size,
                              hipStream_t stream) {
  const float* X    = (const float*)d_in[0];
  const float* tp   = (const float*)d_in[1];
  const float* npd  = (const float*)d_in[2];
  const float* alen = (const float*)d_in[3];
  const float* anch = (const float*)d_in[4];
  auto P = [&](int s, int k) -> const float* { return (const float*)d_in[5 + s * 13 + k]; };
  // k: 0 wih_f, 1 whh_f, 2 bih_f, 3 bhh_f, 4 wih_b, 5 whh_b, 6 bih_b, 7 bhh_b,
  //    8 w1, 9 w2, 10 w3, 11 sw, 12 ew
  static const int BINS_H[3] = {80, 60, 80};

  float* ws = (float*)d_ws;
  float* gx = ws;                                      // 6*512*768
  float* hfin = ws + (size_t)6 * 512 * 768;            // 6*48*256
  unsigned int* scales = (unsigned int*)(hfin + (size_t)6 * 48 * 256);  // 9

  GxArgs ga;
  ga.X = X; ga.gx = gx;
  for (int s = 0; s < 3; ++s) {
    ga.wih[s * 2 + 0] = P(s, 0); ga.bih[s * 2 + 0] = P(s, 2);
    ga.wih[s * 2 + 1] = P(s, 4); ga.bih[s * 2 + 1] = P(s, 6);
  }
  gx_kernel<<<dim3(1152), dim3(256), 0, stream>>>(ga);  // 9216 waves, 1 tile each

  MaxArgs ma;
  for (int s = 0; s < 3; ++s) {
    ma.w[s * 3 + 0] = P(s, 8);  ma.len[s * 3 + 0] = 256 * 515;
    ma.w[s * 3 + 1] = P(s, 9);  ma.len[s * 3 + 1] = 256 * 256;
    ma.w[s * 3 + 2] = P(s, 10); ma.len[s * 3 + 2] = (2 * BINS_H[s] + 5) * 256;
  }
  ma.out = scales;
  maxabs_kernel<<<dim3(9), dim3(256), 0, stream>>>(ma);

  ScanArgs sa;
  for (int s = 0; s < 3; ++s) {
    sa.whh[s * 2 + 0] = P(s, 1); sa.bhh[s * 2 + 0] = P(s, 3);
    sa.whh[s * 2 + 1] = P(s, 5); sa.bhh[s * 2 + 1] = P(s, 7);
  }
  sa.gx = gx; sa.tp = tp; sa.audio_len = alen; sa.anch = anch; sa.hfin = hfin;
  gru_scan_kernel<<<dim3(10), dim3(512), 0, stream>>>(sa);  // 10 row-tile scans

  HeadArgs ha;
  ha.hfin = hfin; ha.scales = scales; ha.node_pred = npd;
  ha.tp = tp; ha.audio_len = alen; ha.anch = anch; ha.out = (float*)d_out;
  for (int s = 0; s < 3; ++s) {
    ha.w1[s] = P(s, 8); ha.w2[s] = P(s, 9); ha.w3[s] = P(s, 10);
    ha.sw[s] = P(s, 11); ha.ew[s] = P(s, 12);
    ha.bih_f[s] = P(s, 2); ha.bhh_f[s] = P(s, 3);
    ha.bih_b[s] = P(s, 6); ha.bhh_b[s] = P(s, 7);
  }
  head_kernel<<<dim3(70), dim3(256), 0, stream>>>(ha);
}